// LocalGrouper_23295902614327
// MI455X (gfx1250) — compile-verified
//
#include <hip/hip_runtime.h>
#include <math.h>

// Problem constants (match reference)
#define BB 4
#define NN 4096
#define DD 256
#define SS_ 1024
#define KK 32
#define C_ALL (DD + 3)      // 259
#define OUT_C (2 * DD + 3)  // 515
#define EPS_ 1e-5f

typedef float v2f __attribute__((ext_vector_type(2)));
typedef float v8f __attribute__((ext_vector_type(8)));

// ---------------------------------------------------------------------------
// Kernel 0: zero per-batch sum-of-squares accumulator (re-run every call so
// graph replay stays deterministic).
// ---------------------------------------------------------------------------
__global__ void lg_init_kernel(float* ss_accum) {
    int t = threadIdx.x;
    if (t < BB) ss_accum[t] = 0.0f;
}

// ---------------------------------------------------------------------------
// Kernel 1: farthest point sampling. One block per batch, sequential over S
// steps, block-wide argmax with first-occurrence tie-break (matches argmax).
// ---------------------------------------------------------------------------
__global__ void lg_fps_kernel(const float* __restrict__ xyz, int* __restrict__ fps_idx) {
    const int b = blockIdx.x;
    const float* X = xyz + (size_t)b * NN * 3;
    __shared__ float dist[NN];       // 16 KB
    __shared__ float red_v[256];
    __shared__ int   red_i[256];
    __shared__ int   cur;
    const int tid = threadIdx.x;     // blockDim.x == 256

    for (int i = tid; i < NN; i += 256) dist[i] = 1e10f;
    if (tid == 0) cur = 0;
    __syncthreads();

    for (int s = 0; s < SS_; ++s) {
        const int far = cur;
        if (tid == 0) fps_idx[b * SS_ + s] = far;
        const float cx = X[far * 3 + 0];
        const float cy = X[far * 3 + 1];
        const float cz = X[far * 3 + 2];
        float best = -1.0f;
        int   bi   = 0;
        for (int i = tid; i < NN; i += 256) {
            const float dx = X[i * 3 + 0] - cx;
            const float dy = X[i * 3 + 1] - cy;
            const float dz = X[i * 3 + 2] - cz;
            const float d  = dx * dx + dy * dy + dz * dz;
            const float dd = fminf(dist[i], d);
            dist[i] = dd;
            if (dd > best) { best = dd; bi = i; }  // strict > keeps earliest i
        }
        red_v[tid] = best;
        red_i[tid] = bi;
        __syncthreads();
        for (int off = 128; off > 0; off >>= 1) {
            if (tid < off) {
                const float v2 = red_v[tid + off];
                const int   i2 = red_i[tid + off];
                if (v2 > red_v[tid] || (v2 == red_v[tid] && i2 < red_i[tid])) {
                    red_v[tid] = v2; red_i[tid] = i2;
                }
            }
            __syncthreads();
        }
        if (tid == 0) cur = red_i[0];
        __syncthreads();
    }
}

// ---------------------------------------------------------------------------
// Kernel 2: KNN via V_WMMA_F32_16X16X4_F32.
// One wave (32 threads) per block handles 16 centroids.
//   A[m][k] = (-2cx, -2cy, -2cz, 1)    (16x4)
//   B[k][n] = ( px,   py,   pz, |p|^2) (4x16)
//   (A.B)[m][n] = |p_n|^2 - 2 c_m . p_n  -> off true sq-dist by a per-row
// constant (|c_m|^2), so per-row top-k ordering matches the reference exactly.
// Fragment construction is branch-free (v_cndmask selects), keeping the hot
// loop free of exec-mask churn; WMMA requires EXEC all-ones anyway.
// A-frag layout: lane<16 holds (K=0,K=1) of row M=lane; lane>=16 holds (K=2,K=3).
// B-frag layout mirrors with N=lane%16.
// C/D layout: lane l, vgpr j -> row j + 8*(l/16), col l%16.
// ---------------------------------------------------------------------------
#define KNN_CH 128  // columns per LDS chunk

__global__ void lg_knn_kernel(const float* __restrict__ xyz,
                              const int* __restrict__ fps_idx,
                              int* __restrict__ knn_idx) {
    const int b    = blockIdx.y;
    const int s0   = blockIdx.x * 16;
    const int lane = threadIdx.x;  // 32 threads = 1 wave
    const float* X = xyz + (size_t)b * NN * 3;
    const bool lo  = (lane < 16);

    __shared__ float dchunk[16][KNN_CH];  // 8 KB
    __shared__ float topv[16][KK];
    __shared__ int   topi[16][KK];
    __shared__ float maxv[16];
    __shared__ int   maxp[16];

    // A fragment (centroid tile), constant across all column tiles
    const int m  = lane & 15;
    const int ci = fps_idx[b * SS_ + s0 + m];
    const float cx = X[ci * 3 + 0];
    const float cy = X[ci * 3 + 1];
    const float cz = X[ci * 3 + 2];
    v2f afrag;
    afrag.x = lo ? (-2.0f * cx) : (-2.0f * cz);
    afrag.y = lo ? (-2.0f * cy) : 1.0f;

    if (lane < 16) {
        for (int j = 0; j < KK; ++j) { topv[lane][j] = 1e30f; topi[lane][j] = 0; }
        maxv[lane] = 1e30f; maxp[lane] = 0;
    }
    __syncthreads();

    for (int c0 = 0; c0 < NN; c0 += KNN_CH) {
        // Prefetch next chunk's coordinates (global_prefetch_b8)
        if (c0 + KNN_CH < NN) {
            __builtin_prefetch(&X[(c0 + KNN_CH + lane * 4) * 3], 0, 1);
        }
        // --- distance tiles via WMMA (all 32 lanes active: EXEC all ones) ---
        for (int t = 0; t < KNN_CH; t += 16) {
            const int n  = c0 + t + (lane & 15);
            const float px = X[n * 3 + 0];
            const float py = X[n * 3 + 1];
            const float pz = X[n * 3 + 2];
            const float sq = px * px + py * py + pz * pz;
            v2f bfrag;
            bfrag.x = lo ? px : pz;   // branch-free: v_cndmask
            bfrag.y = lo ? py : sq;
            v8f cacc = {};
            cacc = __builtin_amdgcn_wmma_f32_16x16x4_f32(
                false, afrag, false, bfrag, (short)0, cacc, false, false);
            const int rbase = lo ? 0 : 8;
            const int col   = t + (lane & 15);
#pragma unroll
            for (int j = 0; j < 8; ++j) dchunk[rbase + j][col] = cacc[j];
        }
        __syncthreads();
        // --- streaming top-K selection, one lane per centroid row ---
        if (lane < 16) {
            float mv = maxv[lane];
            int   mp = maxp[lane];
            for (int cc = 0; cc < KNN_CH; ++cc) {
                const float d = dchunk[lane][cc];
                if (d < mv) {
                    topv[lane][mp] = d;
                    topi[lane][mp] = c0 + cc;
                    mv = topv[lane][0]; mp = 0;
                    for (int j = 1; j < KK; ++j)
                        if (topv[lane][j] > mv) { mv = topv[lane][j]; mp = j; }
                }
            }
            maxv[lane] = mv; maxp[lane] = mp;
        }
        __syncthreads();
    }

    // sort ascending by (distance, index) like top_k(-sq), then emit
    if (lane < 16) {
        for (int i = 1; i < KK; ++i) {
            const float v  = topv[lane][i];
            const int   id = topi[lane][i];
            int j = i - 1;
            while (j >= 0 && (topv[lane][j] > v ||
                              (topv[lane][j] == v && topi[lane][j] > id))) {
                topv[lane][j + 1] = topv[lane][j];
                topi[lane][j + 1] = topi[lane][j];
                --j;
            }
            topv[lane][j + 1] = v;
            topi[lane][j + 1] = id;
        }
        int* out = knn_idx + ((size_t)b * SS_ + s0 + lane) * KK;
        for (int j = 0; j < KK; ++j) out[j] = topi[lane][j];
    }
}

// ---------------------------------------------------------------------------
// Kernel 3: per-(b,s,c) mean over K neighbors + per-batch sum of squared
// deviations: SS += sumsq - sum*mu  (== sum_k (x-mu)^2).
// ---------------------------------------------------------------------------
__global__ void lg_stats_kernel(const float* __restrict__ xyz,
                                const float* __restrict__ f,
                                const int* __restrict__ knn_idx,
                                float* __restrict__ mean,
                                float* __restrict__ ss_accum) {
    const int bs = blockIdx.x;
    const int b  = bs / SS_;
    const int tid = threadIdx.x;  // 256
    __shared__ int   nidx[KK];
    __shared__ float partial[256];
    if (tid < KK) nidx[tid] = knn_idx[(size_t)bs * KK + tid];
    __syncthreads();
    const float* F = f   + (size_t)b * NN * DD;
    const float* X = xyz + (size_t)b * NN * 3;
    float local_ssd = 0.0f;
    for (int c = tid; c < C_ALL; c += 256) {
        float sum = 0.0f, sumsq = 0.0f;
        for (int k = 0; k < KK; ++k) {
            const int n = nidx[k];
            const float v = (c < DD) ? F[(size_t)n * DD + c] : X[n * 3 + (c - DD)];
            sum   += v;
            sumsq += v * v;
        }
        const float mu = sum * (1.0f / KK);
        mean[(size_t)bs * C_ALL + c] = mu;
        local_ssd += sumsq - sum * mu;
    }
    partial[tid] = local_ssd;
    __syncthreads();
    for (int off = 128; off > 0; off >>= 1) {
        if (tid < off) partial[tid] += partial[tid + off];
        __syncthreads();
    }
    if (tid == 0) atomicAdd(&ss_accum[b], partial[0]);
}

// ---------------------------------------------------------------------------
// Kernel 4: std[b] = sqrt(SS / (n-1))   (ddof=1; centered array has zero mean)
// ---------------------------------------------------------------------------
__global__ void lg_std_kernel(const float* __restrict__ ss_accum, float* __restrict__ stdv) {
    int t = threadIdx.x;
    if (t < BB) {
        const float cnt = (float)((size_t)SS_ * KK * C_ALL - 1);
        stdv[t] = sqrtf(ss_accum[t] / cnt);
    }
}

// ---------------------------------------------------------------------------
// Kernel 5: write xyz_sampled and f_out = [(grouped - mean)/(std+eps) || f_sampled]
// ---------------------------------------------------------------------------
__global__ void lg_output_kernel(const float* __restrict__ xyz,
                                 const float* __restrict__ f,
                                 const int* __restrict__ fps_idx,
                                 const int* __restrict__ knn_idx,
                                 const float* __restrict__ mean,
                                 const float* __restrict__ stdv,
                                 float* __restrict__ out_xyz,
                                 float* __restrict__ out_f) {
    const int bs  = blockIdx.x;
    const int b   = bs / SS_;
    const int tid = threadIdx.x;  // 256
    const float* F = f   + (size_t)b * NN * DD;
    const float* X = xyz + (size_t)b * NN * 3;
    __shared__ int   nidx[KK];
    __shared__ float mu[C_ALL];
    __shared__ float fsamp[DD];
    const int ci = fps_idx[bs];
    if (tid < KK) nidx[tid] = knn_idx[(size_t)bs * KK + tid];
    for (int c = tid; c < C_ALL; c += 256) mu[c] = mean[(size_t)bs * C_ALL + c];
    for (int c = tid; c < DD; c += 256) fsamp[c] = F[(size_t)ci * DD + c];
    if (tid < 3) out_xyz[(size_t)bs * 3 + tid] = X[ci * 3 + tid];
    __syncthreads();
    const float inv = 1.0f / (stdv[b] + EPS_);
    float* OB = out_f + (size_t)bs * KK * OUT_C;
    for (int k = 0; k < KK; ++k) {
        const int n = nidx[k];
        float* O = OB + (size_t)k * OUT_C;
        for (int c = tid; c < OUT_C; c += 256) {
            float v;
            if (c < C_ALL) {
                const float x = (c < DD) ? F[(size_t)n * DD + c] : X[n * 3 + (c - DD)];
                v = (x - mu[c]) * inv;
            } else {
                v = fsamp[c - C_ALL];
            }
            O[c] = v;
        }
    }
}

// ---------------------------------------------------------------------------
// Host launcher
// ---------------------------------------------------------------------------
extern "C" void kernel_launch(void* const* d_in, const int* in_sizes, int n_in,
                              void* d_out, int out_size, void* d_ws, size_t ws_size,
                              hipStream_t stream) {
    const float* xyz = (const float*)d_in[0];
    const float* f   = (const float*)d_in[1];
    // d_in[2]/d_in[3] (affine_alpha/beta) do not affect the reference output.

    float* out      = (float*)d_out;
    float* out_xyz  = out;                            // [B,S,3]
    float* out_f    = out + (size_t)BB * SS_ * 3;     // [B,S,K,2D+3]

    char* ws = (char*)d_ws;
    int*   fps_idx = (int*)ws;    ws += (size_t)BB * SS_ * sizeof(int);
    int*   knn_idx = (int*)ws;    ws += (size_t)BB * SS_ * KK * sizeof(int);
    float* mean    = (float*)ws;  ws += (size_t)BB * SS_ * C_ALL * sizeof(float);
    float* ss_acc  = (float*)ws;  ws += (size_t)BB * sizeof(float);
    float* stdv    = (float*)ws;

    lg_init_kernel<<<1, 32, 0, stream>>>(ss_acc);
    lg_fps_kernel<<<BB, 256, 0, stream>>>(xyz, fps_idx);
    lg_knn_kernel<<<dim3(SS_ / 16, BB), 32, 0, stream>>>(xyz, fps_idx, knn_idx);
    lg_stats_kernel<<<BB * SS_, 256, 0, stream>>>(xyz, f, knn_idx, mean, ss_acc);
    lg_std_kernel<<<1, 32, 0, stream>>>(ss_acc, stdv);
    lg_output_kernel<<<BB * SS_, 256, 0, stream>>>(xyz, f, fps_idx, knn_idx, mean, stdv,
                                                   out_xyz, out_f);
}